// ChebConvLayer_78434692759896
// MI455X (gfx1250) — compile-verified
//
#include <hip/hip_runtime.h>

// CDNA5 / gfx1250 ChebConv: out = sum_b T_b @ W_b,  T_k = 2*gso@T_{k-1} - T_{k-2}
// fp32 WMMA (V_WMMA_F32_16X16X4_F32), wave32, TDM-staged B operands, double-buffered.

typedef __attribute__((ext_vector_type(2))) float v2f;
typedef __attribute__((ext_vector_type(4))) float v4f;
typedef __attribute__((ext_vector_type(8))) float v8f;
typedef __attribute__((ext_vector_type(4))) unsigned int u32x4;
typedef __attribute__((ext_vector_type(8))) int i32x8;
typedef __attribute__((ext_vector_type(4))) int i32x4;

#define N_NODES 10000
#define FDIM    32
#define CHEB_ORDER 4
#define KSPLIT  5                    // waves per workgroup (K-split)
#define KSEG    (N_NODES / KSPLIT)   // 2000 (divisible by 8)
#define KT      40                   // K rows staged per TDM transfer (divisible by 8)
#define NCHUNK  (KSEG / KT)          // 50

// Tensor-DMA one [KT x 32] fp32 tile of tprev into LDS (D# per ISA 8.3/8.4).
// This toolchain's builtin: (u32x4 g0, i32x8 g1, i32x4 g2, i32x4 g3, i32x8 g4, i32 cpol)
__device__ __forceinline__ void tdm_load_b(unsigned lds_off,
                                           unsigned long long gaddr,
                                           int nrows) {
    u32x4 g0;
    g0[0] = 1u;                                               // count=1, user mode
    g0[1] = lds_off;                                          // lds_addr
    g0[2] = (unsigned)(gaddr & 0xFFFFFFFFull);                // global_addr[31:0]
    g0[3] = (unsigned)((gaddr >> 32) & 0x01FFFFFFull)         // global_addr[56:32]
            | 0x80000000u;                                    // type=2 ("image")
    i32x8 g1;
    g1[0] = 0x00020000;                // workgroup_mask=0, data_size=2 (4 bytes)
    g1[1] = (int)(32u << 16);          // tensor_dim0 = 32 (bits 63:48)
    g1[2] = (int)(10000u << 16);       // tensor_dim1[15:0] = 10000 (bits 95:80)
    g1[3] = (int)(32u << 16);          // tile_dim0 = 32 (bits 127:112)
    g1[4] = nrows;                     // tile_dim1 (bits 143:128)
    g1[5] = 32;                        // tensor_dim0_stride = 32 (bits 191:160)
    g1[6] = 0;
    g1[7] = 0;
    i32x4 g2 = {0, 0, 0, 0};
    i32x4 g3 = {0, 0, 0, 0};
    i32x8 g4 = {0, 0, 0, 0, 0, 0, 0, 0};
    __builtin_amdgcn_tensor_load_to_lds(g0, g1, g2, g3, g4, 0);
}

// One workgroup = one 16-row output block x all 32 cols. 5 waves split K
// (2000 each); each wave TDM-stages its own B chunks (no cross-wave sync,
// s_wait_tensorcnt only); LDS tree-reduction at the end (deterministic).
__global__ __launch_bounds__(32 * KSPLIT) void cheb_spectral_pass(
    const float* __restrict__ gso,    // [N, N] row-major
    const float* __restrict__ tprev,  // [N, 32]
    const float* __restrict__ tsub,   // [N, 32] (T_{k-2}); ignored if beta==0
    float* __restrict__ tout,         // [N, 32]
    float alpha, float beta)
{
    __shared__ float bstage[KSPLIT][2][KT * FDIM];   // 5*2*5120B = 51200B
    __shared__ float red[KSPLIT][16][FDIM];          // 10240B

    const int tid  = threadIdx.x;
    const int wave = __builtin_amdgcn_readfirstlane(tid >> 5);  // 0..4, uniform
    const int lane = tid & 31;
    const int hi   = lane >> 4;       // lane half
    const int c    = lane & 15;       // row (A) / col (B,C) in tile
    const int mbase = blockIdx.x * 16;
    const int k0 = wave * KSEG;

    // A (16x4 f32 layout): both lane halves hold rows 0-15; a float4 at col
    // k+4*hi supplies TWO K-steps via slot permutation (see B row pairing).
    const float* arow = gso + (size_t)(mbase + c) * N_NODES + 4 * hi;

    const unsigned bst_off =
        (unsigned)(size_t)(&bstage[0][0][0]) + (unsigned)(wave * 2 * KT * FDIM * 4);
    const unsigned long long gbase =
        (unsigned long long)(size_t)tprev + (unsigned long long)k0 * FDIM * 4u;

    v8f acc0 = {};   // cols 0..15
    v8f acc1 = {};   // cols 16..31

    tdm_load_b(bst_off, gbase, KT);                         // prologue: chunk 0

    for (int ch = 0; ch < NCHUNK; ++ch) {
        if (ch + 1 < NCHUNK) {
            // buffer (ch+1)&1 was last read in iteration ch-1; fence ds reads
            asm volatile("s_wait_dscnt 0" ::: "memory");
            tdm_load_b(bst_off + (unsigned)(((ch + 1) & 1) * KT * FDIM * 4),
                       gbase + (unsigned long long)(ch + 1) * KT * FDIM * 4u, KT);
            __builtin_amdgcn_s_wait_tensorcnt(1);           // chunk ch complete
        } else {
            __builtin_amdgcn_s_wait_tensorcnt(0);
        }
        asm volatile("" ::: "memory");

        const float* bsh = &bstage[wave][ch & 1][0];
        const int kg = k0 + ch * KT;
        __builtin_prefetch(arow + kg + 4 * KT, 0, 0);       // gso stream ahead

#pragma unroll
        for (int kk = 0; kk < KT; kk += 8) {
            v4f av = *(const v4f*)(arow + kg + kk);
            const float* bb = bsh + (kk + 4 * hi) * FDIM + c;
            // step a: K slots = rows {kk+4hi, kk+4hi+1 | per half}
            v2f a0; a0.x = av.x; a0.y = av.y;
            v2f b0; b0.x = bb[0];  b0.y = bb[FDIM];
            v2f b1; b1.x = bb[16]; b1.y = bb[FDIM + 16];
            acc0 = __builtin_amdgcn_wmma_f32_16x16x4_f32(false, a0, false, b0,
                                                         (short)0, acc0, false, false);
            acc1 = __builtin_amdgcn_wmma_f32_16x16x4_f32(false, a0, false, b1,
                                                         (short)0, acc1, false, false);
            // step b: K slots = rows {kk+4hi+2, kk+4hi+3 | per half}
            v2f a1; a1.x = av.z; a1.y = av.w;
            v2f c0; c0.x = bb[2 * FDIM];      c0.y = bb[3 * FDIM];
            v2f c1; c1.x = bb[2 * FDIM + 16]; c1.y = bb[3 * FDIM + 16];
            acc0 = __builtin_amdgcn_wmma_f32_16x16x4_f32(false, a1, false, c0,
                                                         (short)0, acc0, false, false);
            acc1 = __builtin_amdgcn_wmma_f32_16x16x4_f32(false, a1, false, c1,
                                                         (short)0, acc1, false, false);
        }
    }

    // C/D layout: VGPR j = rows {j | j+8}, lanes = cols.
#pragma unroll
    for (int j = 0; j < 8; ++j) {
        red[wave][j + 8 * hi][c]      = acc0[j];
        red[wave][j + 8 * hi][c + 16] = acc1[j];
    }
    __syncthreads();

    // Fixed-order reduction of 5 partials + affine combine (deterministic).
    for (int e = tid; e < 16 * FDIM; e += 32 * KSPLIT) {
        const int r = e >> 5, col = e & 31;
        float v = red[0][r][col] + red[1][r][col] + red[2][r][col] +
                  red[3][r][col] + red[4][r][col];
        float o = alpha * v;
        if (beta != 0.0f)
            o += beta * tsub[(size_t)(mbase + r) * FDIM + col];
        tout[(size_t)(mbase + r) * FDIM + col] = o;
    }
}

// out[16x32 tile] = sum_b T_b[tile] @ W_b   (K=32 per order, all via WMMA)
__global__ __launch_bounds__(256) void cheb_weight_contract(
    const float* __restrict__ t0, const float* __restrict__ t1,
    const float* __restrict__ t2, const float* __restrict__ t3,
    const float* __restrict__ w,   // [ORDER, 32, 32]
    float* __restrict__ out)       // [N, 32]
{
    const int lane = threadIdx.x & 31;
    const int wav  = threadIdx.x >> 5;
    const int tile = blockIdx.x * 8 + wav;
    if (tile >= N_NODES / 16) return;   // whole-wave uniform exit

    const int hi = lane >> 4, c = lane & 15;
    const int mbase = tile * 16;
    const float* tsrc[CHEB_ORDER] = {t0, t1, t2, t3};

    v8f acc0 = {}, acc1 = {};
#pragma unroll
    for (int b = 0; b < CHEB_ORDER; ++b) {
        const float* ap = tsrc[b] + (size_t)(mbase + c) * FDIM + 2 * hi;
        const float* wb = w + b * FDIM * FDIM;
#pragma unroll
        for (int k = 0; k < FDIM; k += 4) {
            v2f a = *(const v2f*)(ap + k);
            const float* bb = wb + (k + 2 * hi) * FDIM + c;
            v2f b0; b0.x = bb[0];  b0.y = bb[FDIM];
            v2f b1; b1.x = bb[16]; b1.y = bb[FDIM + 16];
            acc0 = __builtin_amdgcn_wmma_f32_16x16x4_f32(false, a, false, b0,
                                                         (short)0, acc0, false, false);
            acc1 = __builtin_amdgcn_wmma_f32_16x16x4_f32(false, a, false, b1,
                                                         (short)0, acc1, false, false);
        }
    }
#pragma unroll
    for (int j = 0; j < 8; ++j) {
        out[(size_t)(mbase + j + 8 * hi) * FDIM + c]      = acc0[j];
        out[(size_t)(mbase + j + 8 * hi) * FDIM + c + 16] = acc1[j];
    }
}

extern "C" void kernel_launch(void* const* d_in, const int* in_sizes, int n_in,
                              void* d_out, int out_size, void* d_ws, size_t ws_size,
                              hipStream_t stream) {
    (void)in_sizes; (void)n_in; (void)out_size; (void)ws_size;

    const float* x   = (const float*)d_in[0];   // [N, 32]
    const float* gso = (const float*)d_in[1];   // [N, N]
    const float* w   = (const float*)d_in[2];   // [4, 32, 32]
    float* out = (float*)d_out;                 // [N, 32]

    float* T1 = (float*)d_ws;                   // 3 * N*32 fp32 scratch
    float* T2 = T1 + (size_t)N_NODES * FDIM;
    float* T3 = T2 + (size_t)N_NODES * FDIM;

    const dim3 grid(N_NODES / 16), block(32 * KSPLIT);
    // T1 = gso @ x
    cheb_spectral_pass<<<grid, block, 0, stream>>>(gso, x,  x,  T1, 1.0f,  0.0f);
    // T2 = 2*gso@T1 - x
    cheb_spectral_pass<<<grid, block, 0, stream>>>(gso, T1, x,  T2, 2.0f, -1.0f);
    // T3 = 2*gso@T2 - T1
    cheb_spectral_pass<<<grid, block, 0, stream>>>(gso, T2, T1, T3, 2.0f, -1.0f);
    // out = x@W0 + T1@W1 + T2@W2 + T3@W3
    cheb_weight_contract<<<(N_NODES / 16 + 7) / 8, 256, 0, stream>>>(
        x, T1, T2, T3, w, out);
}